// GCN2_781684048051
// MI455X (gfx1250) — compile-verified
//
#include <hip/hip_runtime.h>

// ---------------------------------------------------------------------------
// GCN layer on MI455X (gfx1250):
//   support = x @ W            (fp32 WMMA 16x16x4 — exact fp32 path)
//   agg     = segment_sum(support[src] * w, dst)   (hw f32 atomics)
//   out     = relu(agg + b) + x
// ---------------------------------------------------------------------------

typedef __attribute__((ext_vector_type(2))) float v2f;
typedef __attribute__((ext_vector_type(8))) float v8f;

#define DIMS 512  // feature dim D (not checked against in_sizes at launch)

// ---- kernel 0: zero the atomic accumulator (d_out) ------------------------
__global__ void zero_f4(float4* __restrict__ p, long n4) {
    long i = (long)blockIdx.x * blockDim.x + threadIdx.x;
    if (i < n4) p[i] = make_float4(0.f, 0.f, 0.f, 0.f);
}

// ---- kernel 1: support = X @ W via V_WMMA_F32_16X16X4_F32 -----------------
// One wave -> one 16x16 tile of support. 8 waves/block share 16 rows of X.
// f32 A (16x4) layout: lanes 0-15 hold {K=0,K=1}, lanes 16-31 hold {K=2,K=3}.
// f32 B (4x16) layout mirrors A; C/D: lane<16 rows M=vgpr, lane>=16 rows M=vgpr+8.
__launch_bounds__(256)
__global__ void gemm_xw_wmma(const float* __restrict__ X,
                             const float* __restrict__ W,
                             float* __restrict__ S,
                             int nRows) {
    const int wave = threadIdx.x >> 5;
    const int lane = threadIdx.x & 31;
    const int half = lane >> 4;     // 0: K=0,1 / rows 0..7 ; 1: K=2,3 / rows 8..15
    const int l    = lane & 15;

    const int col0 = (blockIdx.x * 8 + wave) * 16;  // N-tile (0..31)
    const int row0 = blockIdx.y * 16;               // M-tile
    if (row0 >= nRows) return;

    const float* __restrict__ xrow = X + (size_t)(row0 + l) * DIMS;
    const float* __restrict__ wcol = W + col0 + l;

    v8f c = {};
    #pragma unroll 4
    for (int k0 = 0; k0 < DIMS; k0 += 4) {
        const int kk = k0 + 2 * half;                 // even -> 8B aligned
        v2f a = *(const v2f*)(xrow + kk);             // A frag: K=kk, kk+1
        v2f b;
        b.x = wcol[(size_t)kk * DIMS];                // B frag: W[kk , col]
        b.y = wcol[(size_t)(kk + 1) * DIMS];          //         W[kk+1, col]
        c = __builtin_amdgcn_wmma_f32_16x16x4_f32(
                /*neg_a=*/false, a, /*neg_b=*/false, b,
                /*c_mod=*/(short)0, c, /*reuse_a=*/false, /*reuse_b=*/false);
    }

    float* __restrict__ srow = S + (size_t)(row0 + half * 8) * DIMS + col0 + l;
    #pragma unroll
    for (int i = 0; i < 8; ++i)
        srow[(size_t)i * DIMS] = c[i];
}

// ---- kernel 2: per-edge scatter: out[dst] += support[src] * w -------------
// One edge per 128-thread block; float4 coalesced row read, 4 hw f32 atomics.
__launch_bounds__(128)
__global__ void edge_scatter(const float* __restrict__ S,
                             const float* __restrict__ ew,
                             const int* __restrict__ esrc,
                             const int* __restrict__ edst,
                             float* __restrict__ out) {
    const int e = blockIdx.x;
    const float w  = ew[e];                    // scalar (uniform) loads
    const size_t s = (size_t)esrc[e] * DIMS;
    const size_t t = (size_t)edst[e] * DIMS;
    const int d = threadIdx.x * 4;             // 128 threads x 4 floats = 512
    const float4 v = *(const float4*)(S + s + d);
    float* o = out + t + d;
    unsafeAtomicAdd(o + 0, v.x * w);           // global_atomic_add_f32
    unsafeAtomicAdd(o + 1, v.y * w);
    unsafeAtomicAdd(o + 2, v.z * w);
    unsafeAtomicAdd(o + 3, v.w * w);
}

// ---- kernel 3: out = relu(out + b) + x  (in place, float4) ----------------
__global__ void finalize_f4(float* __restrict__ out,
                            const float* __restrict__ X,
                            const float* __restrict__ b,
                            long n4) {
    long i = (long)blockIdx.x * blockDim.x + threadIdx.x;
    if (i >= n4) return;
    const float4 agg = ((const float4*)out)[i];
    const float4 x   = ((const float4*)X)[i];
    const int col4   = (int)(i & (DIMS / 4 - 1)) * 4;
    const float4 bb  = *(const float4*)(b + col4);
    float4 r;
    r.x = fmaxf(agg.x + bb.x, 0.f) + x.x;
    r.y = fmaxf(agg.y + bb.y, 0.f) + x.y;
    r.z = fmaxf(agg.z + bb.z, 0.f) + x.z;
    r.w = fmaxf(agg.w + bb.w, 0.f) + x.w;
    ((float4*)out)[i] = r;
}

extern "C" void kernel_launch(void* const* d_in, const int* in_sizes, int n_in,
                              void* d_out, int out_size, void* d_ws, size_t ws_size,
                              hipStream_t stream) {
    const float* X    = (const float*)d_in[0];  // [N, D]
    const float* W    = (const float*)d_in[1];  // [D, D]
    const float* b    = (const float*)d_in[2];  // [D]
    const float* ew   = (const float*)d_in[3];  // [E]
    const int*   esrc = (const int*)d_in[4];    // [E]
    const int*   edst = (const int*)d_in[5];    // [E]
    float*       out  = (float*)d_out;          // [N, D]
    float*       S    = (float*)d_ws;           // scratch: support [N, D]

    const int Dd = in_sizes[2];                 // 512
    const int N  = in_sizes[0] / Dd;            // 50000
    const int E  = in_sizes[3];                 // 1600000
    (void)n_in; (void)out_size; (void)ws_size; (void)Dd;

    const long n4 = (long)N * DIMS / 4;

    // 0) zero atomic accumulator (d_out is poisoned; must be deterministic)
    zero_f4<<<(unsigned)((n4 + 255) / 256), 256, 0, stream>>>((float4*)out, n4);

    // 1) support = X @ W   (grid: 4 col-tile groups x 3125 row tiles)
    dim3 gg(DIMS / 16 / 8, (unsigned)((N + 15) / 16));
    gemm_xw_wmma<<<gg, 256, 0, stream>>>(X, W, S, N);

    // 2) scatter-add all edges
    edge_scatter<<<(unsigned)E, DIMS / 4, 0, stream>>>(S, ew, esrc, edst, out);

    // 3) out = relu(out + b) + X
    finalize_f4<<<(unsigned)((n4 + 255) / 256), 256, 0, stream>>>(out, X, b, n4);
}